// interactionModule_10599979287287
// MI455X (gfx1250) — compile-verified
//
#include <hip/hip_runtime.h>
#include <hip/hip_bf16.h>

// ---------------------------------------------------------------------------
// interactionModule: Kuramoto-style edge interaction.
//   v[n]      = v0 * (cos theta_n, sin theta_n)
//   msum[d]  += (cos(th_s - th_d), sin(th_s - th_d))   over edges (s,d)
//   torque[n] = w0 * msum.y / max(|msum|, 1e-12)   (count cancels in normalize)
//
// CDNA5 specifics used:
//   * tensor_load_to_lds (TDM) double-buffered: streams contiguous edge-index
//     tiles into LDS, overlapping DMA with compute and freeing the VMEM pipe
//     for the random theta gathers + f32 scatter atomics (both L2-resident).
//   * s_wait_tensorcnt(2)/(0) for in-order TDM pipelining.
//   * hardware global_atomic_add_f32 (no-return) via unsafeAtomicAdd.
// ---------------------------------------------------------------------------

#define TILE 2048          // edges per tile; 2 buffers x (src+dst) = 32 KB LDS
#define EDGE_BLOCK 256     // 8 waves (wave32)

typedef unsigned int v4u __attribute__((ext_vector_type(4)));
typedef int          v8i __attribute__((ext_vector_type(8)));
typedef int          v4i __attribute__((ext_vector_type(4)));

__device__ __forceinline__ unsigned lds_addr(const void* p) {
  // Flat LDS addresses carry the byte offset in the low 32 bits.
  return (unsigned)(size_t)p;
}

// Issue one TDM 1-D copy: count x 4-byte elements, global -> LDS.
__device__ __forceinline__ void tdm_copy_to_lds(const void* gptr,
                                                unsigned lds_byte_addr,
                                                int count) {
  unsigned long long ga = (unsigned long long)gptr;

  // D# group 0 (128b): [1:0]=#valid descriptors=1, [63:32]=lds_addr,
  // [120:64]=global_addr(57b), [127:126]=type=2 ("image").
  v4u g0;
  g0[0] = 1u;
  g0[1] = lds_byte_addr;
  g0[2] = (unsigned)(ga & 0xFFFFFFFFu);
  g0[3] = (unsigned)((ga >> 32) & 0x1FFFFFFu) | (2u << 30);

  // D# group 1 (256b): workgroup_mask=0 (not in cluster), data_size=2 (4B),
  // tensor_dim0 = count (bits 79:48), tensor_dim1 = 1 (bits 111:80),
  // tile_dim0 = count (bits 127:112), tile_dim1/2 = 0 (1-D tile).
  v8i g1;
  g1[0] = (2 << 16);                       // data_size = 4 bytes
  g1[1] = (count & 0xFFFF) << 16;          // tensor_dim0[15:0]
  g1[2] = ((count >> 16) & 0xFFFF)         // tensor_dim0[31:16]
        | (1 << 16);                       // tensor_dim1[15:0] = 1
  g1[3] = (count & 0xFFFF) << 16;          // tile_dim0 (tensor_dim1 hi = 0)
  g1[4] = 0;                               // tile_dim1 = tile_dim2 = 0
  g1[5] = 0;                               // tensor_dim0_stride (unused)
  g1[6] = 0;
  g1[7] = 0;

  v4i z4 = (v4i)0;                         // groups 2/3 unused (<=2-D tensor)
  v8i z8 = (v8i)0;
  __builtin_amdgcn_tensor_load_to_lds(g0, g1, z4, z4, z8, /*cpol=*/0);
}

// ---- Kernel 1: per-node velocity output + zero the edge accumulator -------
__global__ __launch_bounds__(256) void node_v_kernel(
    const float* __restrict__ theta, const float* __restrict__ v0p,
    float* __restrict__ out_v, float* __restrict__ acc, int n_nodes) {
  int n = blockIdx.x * blockDim.x + threadIdx.x;
  if (n < n_nodes) {
    float s, c;
    sincosf(theta[n], &s, &c);   // accurate libm: feeds validated output v
    float v0 = *v0p;
    out_v[2 * n + 0] = v0 * c;
    out_v[2 * n + 1] = v0 * s;
    acc[2 * n + 0] = 0.0f;
    acc[2 * n + 1] = 0.0f;
  }
}

// ---- Kernel 2: edge scatter-reduce, TDM double-buffered index tiles -------
__global__ __launch_bounds__(EDGE_BLOCK) void edge_kernel(
    const int* __restrict__ src, const int* __restrict__ dst,
    const float* __restrict__ theta, float* __restrict__ acc, int n_edges) {
  __shared__ int lsrc[2][TILE];
  __shared__ int ldst[2][TILE];

  const int tid = threadIdx.x;
  const int ntiles = (n_edges + TILE - 1) / TILE;

  int tile = blockIdx.x;
  if (tile >= ntiles) return;

  // Prologue: wave 0 kicks off DMA of this block's first tile into buffer 0.
  if (tid < 32) {
    int base = tile * TILE;
    int rem = n_edges - base;
    int cnt = rem < TILE ? rem : TILE;
    tdm_copy_to_lds(src + base, lds_addr(lsrc[0]), cnt);
    tdm_copy_to_lds(dst + base, lds_addr(ldst[0]), cnt);
  }

  int buf = 0;
  for (; tile < ntiles; tile += gridDim.x) {
    const int base = tile * TILE;
    int rem = n_edges - base;
    const int count = rem < TILE ? rem : TILE;
    const int next = tile + gridDim.x;

    if (tid < 32) {
      if (next < ntiles) {
        // Prefetch next tile into the other buffer (read last iteration,
        // protected by the trailing barrier), then wait for current tile:
        // TDM ops retire in-order per wave, so tensorcnt<=2 => first 2 done.
        int nbase = next * TILE;
        int nrem = n_edges - nbase;
        int ncnt = nrem < TILE ? nrem : TILE;
        tdm_copy_to_lds(src + nbase, lds_addr(lsrc[buf ^ 1]), ncnt);
        tdm_copy_to_lds(dst + nbase, lds_addr(ldst[buf ^ 1]), ncnt);
        __builtin_amdgcn_s_wait_tensorcnt(2);
      } else {
        __builtin_amdgcn_s_wait_tensorcnt(0);
      }
    }
    __syncthreads();  // current tile visible to all 8 waves

    for (int i = tid; i < count; i += EDGE_BLOCK) {
      int s = lsrc[buf][i];
      int d = ldst[buf][i];
      float dth = theta[s] - theta[d];  // D_COEF == 1
      float sn, cs;
      __sincosf(dth, &sn, &cs);  // HW v_sin/v_cos: ~1e-6 abs err, |dth|<~10;
                                 // averaged over ~32 edges then normalized.
      unsafeAtomicAdd(&acc[2 * d + 0], cs);  // global_atomic_add_f32
      unsafeAtomicAdd(&acc[2 * d + 1], sn);
    }
    __syncthreads();  // all reads done before this buffer is DMA-refilled
    buf ^= 1;
  }
}

// ---- Kernel 3: normalize accumulated mailbox, emit torque -----------------
__global__ __launch_bounds__(256) void torque_kernel(
    const float* __restrict__ acc, const float* __restrict__ w0p,
    float* __restrict__ out_t, int n_nodes) {
  int n = blockIdx.x * blockDim.x + threadIdx.x;
  if (n < n_nodes) {
    float sx = acc[2 * n + 0];
    float sy = acc[2 * n + 1];
    // mean = msum/c; unit = mean/max(|mean|,eps) == msum/max(|msum|, c*eps):
    // the count cancels under normalization; cnt==0 => msum==0 => torque 0.
    float norm = sqrtf(sx * sx + sy * sy);
    float denom = fmaxf(norm, 1e-12f);
    out_t[n] = (*w0p) * (sy / denom);
  }
}

extern "C" void kernel_launch(void* const* d_in, const int* in_sizes, int n_in,
                              void* d_out, int out_size, void* d_ws, size_t ws_size,
                              hipStream_t stream) {
  const float* theta = (const float*)d_in[0];
  const int*   src   = (const int*)d_in[1];
  const int*   dst   = (const int*)d_in[2];
  const float* v0    = (const float*)d_in[3];
  const float* w0    = (const float*)d_in[4];

  float* out = (float*)d_out;          // [v (2N floats) | torque (N floats)]
  const int n_nodes = in_sizes[0];     // theta is [N,1]
  const int n_edges = in_sizes[1];

  float* acc = (float*)d_ws;           // 2*N floats (4 MB) — L2 resident

  int nblk = (n_nodes + 255) / 256;
  node_v_kernel<<<nblk, 256, 0, stream>>>(theta, v0, out, acc, n_nodes);

  int ntiles = (n_edges + TILE - 1) / TILE;
  int eblk = ntiles < 16384 ? ntiles : 16384;
  if (eblk < 1) eblk = 1;
  edge_kernel<<<eblk, EDGE_BLOCK, 0, stream>>>(src, dst, theta, acc, n_edges);

  torque_kernel<<<nblk, 256, 0, stream>>>(acc, w0,
                                          out + 2 * (size_t)n_nodes, n_nodes);
}